// intra_message_45930380264258
// MI455X (gfx1250) — compile-verified
//
#include <hip/hip_runtime.h>
#include <math.h>

typedef __attribute__((ext_vector_type(2))) float v2f;
typedef __attribute__((ext_vector_type(8))) float v8f;

#define NN 50000
#define EE 800000
#define DD 128
#define GG 128
#define LDS_STRIDE 132   // 128 + 4 pad: row r, col k -> bank (4r + k) % 64, conflict-free

// ---------------------------------------------------------------------------
// Kernel 1: hv = leaky_relu(node_feats @ W_node + b_node) via f32 WMMA.
// Block = 256 threads = 8 waves; block handles a 16-row node tile, each wave
// one 16-column tile of G. A tile staged in LDS (padded), B read from global
// (64 KB W_node lives in L2/WGP$ after first touch).
// ---------------------------------------------------------------------------
__global__ void __launch_bounds__(256)
node_gemm_wmma(const float* __restrict__ nf, const float* __restrict__ Wn,
               const float* __restrict__ bn, float* __restrict__ hv) {
  __shared__ float Atile[16 * LDS_STRIDE];
  const int row0 = blockIdx.x * 16;
  const int tid  = threadIdx.x;

  // cooperative load: 16x128 floats, 8 per thread, coalesced
  #pragma unroll
  for (int i = tid; i < 16 * DD; i += 256) {
    const int r = i >> 7;
    const int c = i & (DD - 1);
    Atile[r * LDS_STRIDE + c] = nf[(size_t)(row0 + r) * DD + c];
  }
  __syncthreads();

  const int wave = tid >> 5;
  const int lane = tid & 31;
  const int col0 = wave * 16;
  const int half = lane >> 4;   // 0: lanes 0-15, 1: lanes 16-31
  const int l16  = lane & 15;

  v8f acc = {};
  #pragma unroll
  for (int k = 0; k < DD; k += 4) {
    const int ka = k + half * 2;           // lanes 0-15: K=k,k+1 ; 16-31: K=k+2,k+3
    v2f a, b;
    a.x = Atile[l16 * LDS_STRIDE + ka];
    a.y = Atile[l16 * LDS_STRIDE + ka + 1];
    b.x = Wn[(size_t)(ka + 0) * GG + col0 + l16];
    b.y = Wn[(size_t)(ka + 1) * GG + col0 + l16];
    acc = __builtin_amdgcn_wmma_f32_16x16x4_f32(false, a, false, b,
                                                (short)0, acc, false, false);
  }

  // C/D layout: vgpr i -> M = i + 8*half, N = l16
  #pragma unroll
  for (int i = 0; i < 8; ++i) {
    const int row = row0 + i + half * 8;
    const int col = col0 + l16;
    float v = acc[i] + bn[col];
    v = v > 0.0f ? v : 0.01f * v;          // leaky_relu(0.01)
    hv[(size_t)row * GG + col] = v;
  }
}

// ---------------------------------------------------------------------------
// Kernel 2: per-node edge-score halves p[v] = <nf[v], We[0:128]>,
//           q[v] = <nf[v], We[128:256]>. One wave per node.
// ---------------------------------------------------------------------------
__global__ void __launch_bounds__(256)
node_scores(const float* __restrict__ nf, const float* __restrict__ We,
            float* __restrict__ p, float* __restrict__ q) {
  const int wid  = (int)((blockIdx.x * (size_t)blockDim.x + threadIdx.x) >> 5);
  const int lane = threadIdx.x & 31;
  if (wid >= NN) return;
  const float* row = nf + (size_t)wid * DD;
  float ap = 0.0f, aq = 0.0f;
  #pragma unroll
  for (int i = lane; i < DD; i += 32) {
    const float x = row[i];
    ap += x * We[i];
    aq += x * We[DD + i];
  }
  #pragma unroll
  for (int off = 16; off > 0; off >>= 1) {
    ap += __shfl_xor(ap, off, 32);
    aq += __shfl_xor(aq, off, 32);
  }
  if (lane == 0) { p[wid] = ap; q[wid] = aq; }
}

// ---------------------------------------------------------------------------
// Kernel 3: init m=-inf, s=0, out=0  (one pass over N*G elements)
// ---------------------------------------------------------------------------
__global__ void __launch_bounds__(256)
init_buffers(float* __restrict__ m, float* __restrict__ s,
             float* __restrict__ out) {
  const int i = (int)(blockIdx.x * (size_t)blockDim.x + threadIdx.x);
  if (i < NN * GG) out[i] = 0.0f;
  if (i < NN) { m[i] = -INFINITY; s[i] = 0.0f; }
}

// exact float atomic-max via the signed/unsigned bit trick
__device__ inline void atomicMaxF(float* addr, float v) {
  if (v >= 0.0f) atomicMax((int*)addr, __float_as_int(v));
  else           atomicMin((unsigned int*)addr, __float_as_uint(v));
}

// ---------------------------------------------------------------------------
// Kernel 4: edge logits + segment max over dst
// ---------------------------------------------------------------------------
__global__ void __launch_bounds__(256)
edge_logits(const int* __restrict__ src, const int* __restrict__ dst,
            const float* __restrict__ p, const float* __restrict__ q,
            const float* __restrict__ be, float* __restrict__ logit,
            float* __restrict__ m) {
  const int e = (int)(blockIdx.x * (size_t)blockDim.x + threadIdx.x);
  if (e >= EE) return;
  const int d = dst[e];
  float x = p[d] + q[src[e]] + be[0];
  x = x > 0.0f ? x : 0.01f * x;            // leaky_relu(0.01)
  logit[e] = x;
  atomicMaxF(&m[d], x);
}

// ---------------------------------------------------------------------------
// Kernel 5: segment sum of exp(logit - m[dst])
// ---------------------------------------------------------------------------
__global__ void __launch_bounds__(256)
edge_expsum(const int* __restrict__ dst, const float* __restrict__ logit,
            const float* __restrict__ m, float* __restrict__ s) {
  const int e = (int)(blockIdx.x * (size_t)blockDim.x + threadIdx.x);
  if (e >= EE) return;
  const int d = dst[e];
  atomicAdd(&s[d], expf(logit[e] - m[d]));
}

// ---------------------------------------------------------------------------
// Kernel 6: aggregation. One wave per edge: 32 lanes x float4 = 128 floats of
// hv[src], scaled by a_e, atomically accumulated into out[dst]. hv fits in L2.
// ---------------------------------------------------------------------------
__global__ void __launch_bounds__(256)
edge_aggregate(const int* __restrict__ src, const int* __restrict__ dst,
               const float* __restrict__ logit, const float* __restrict__ m,
               const float* __restrict__ s, const float* __restrict__ hv,
               float* __restrict__ out) {
  const int e = (int)((blockIdx.x * (size_t)blockDim.x + threadIdx.x) >> 5);
  const int lane = threadIdx.x & 31;
  if (e >= EE) return;
  const int sv = src[e];
  const int dv = dst[e];
  const float den = s[dv];
  const float a = expf(logit[e] - m[dv]) / (den > 0.0f ? den : 1.0f);
  const float4 h = ((const float4*)(hv + (size_t)sv * GG))[lane];
  float* orow = out + (size_t)dv * GG + lane * 4;
  atomicAdd(orow + 0, h.x * a);
  atomicAdd(orow + 1, h.y * a);
  atomicAdd(orow + 2, h.z * a);
  atomicAdd(orow + 3, h.w * a);
}

// ---------------------------------------------------------------------------
// Kernel 7: in-place elu on the accumulated output
// ---------------------------------------------------------------------------
__global__ void __launch_bounds__(256)
elu_inplace(float* __restrict__ out) {
  const int i = (int)(blockIdx.x * (size_t)blockDim.x + threadIdx.x);
  if (i >= NN * GG) return;
  const float c = out[i];
  out[i] = c > 0.0f ? c : expm1f(c);
}

// ---------------------------------------------------------------------------
extern "C" void kernel_launch(void* const* d_in, const int* in_sizes, int n_in,
                              void* d_out, int out_size, void* d_ws, size_t ws_size,
                              hipStream_t stream) {
  (void)in_sizes; (void)n_in; (void)out_size; (void)ws_size;
  const float* nf  = (const float*)d_in[0];   // [N, D]
  const int*   src = (const int*)  d_in[1];   // [E]
  const int*   dst = (const int*)  d_in[2];   // [E]
  const float* We  = (const float*)d_in[3];   // [2D]
  const float* be  = (const float*)d_in[4];   // [1]
  const float* Wn  = (const float*)d_in[5];   // [D, G]
  const float* bn  = (const float*)d_in[6];   // [G]
  float* out = (float*)d_out;                 // [N, G]

  // workspace layout (floats)
  float* ws    = (float*)d_ws;
  float* hv    = ws;                          // N*G
  float* p     = hv + (size_t)NN * GG;        // N
  float* q     = p + NN;                      // N
  float* m     = q + NN;                      // N
  float* ssum  = m + NN;                      // N
  float* logit = ssum + NN;                   // E

  const int ngBlocks = (NN * GG + 255) / 256; // 25000

  init_buffers<<<ngBlocks, 256, 0, stream>>>(m, ssum, out);
  node_gemm_wmma<<<NN / 16, 256, 0, stream>>>(nf, Wn, bn, hv);
  node_scores<<<(NN * 32 + 255) / 256, 256, 0, stream>>>(nf, We, p, q);
  edge_logits<<<(EE + 255) / 256, 256, 0, stream>>>(src, dst, p, q, be, logit, m);
  edge_expsum<<<(EE + 255) / 256, 256, 0, stream>>>(dst, logit, m, ssum);
  edge_aggregate<<<(EE * 32 + 255) / 256, 256, 0, stream>>>(src, dst, logit, m, ssum, hv, out);
  elu_inplace<<<ngBlocks, 256, 0, stream>>>(out);
}